// HierDDts_63410897158386
// MI455X (gfx1250) — compile-verified
//
#include <hip/hip_runtime.h>
#include <math.h>

typedef float v2f __attribute__((ext_vector_type(2)));
typedef float v8f __attribute__((ext_vector_type(8)));

#define B_   8
#define S_   1024
#define DIN_ 64
#define D_   96
#define NB_  8
#define TWO_PI_F 6.28318530717958647692f
#define LN_EPS_F 1e-5f

// ---------------------------------------------------------------------------
// CDNA5 async global->LDS copy (ASYNCcnt path). Generic pointers to LDS carry
// the LDS byte offset in their low 32 bits,
// so truncation yields the DS address the instruction wants.
// ---------------------------------------------------------------------------
__device__ __forceinline__ void async_ld_f32(float* lds_dst, const float* gsrc) {
    asm volatile("global_load_async_to_lds_b32 %0, %1, off"
                 :
                 : "v"((unsigned)(unsigned long long)(uintptr_t)lds_dst),
                   "v"((unsigned long long)(uintptr_t)gsrc)
                 : "memory");
}
__device__ __forceinline__ void async_wait0() {
    asm volatile("s_wait_asynccnt 0" ::: "memory");
}

// ---------------------------------------------------------------------------
// Kernel A: xt = x @ M^T (M is (96,K)); xn = LN(xt)*g + b
//           optional: res_out = x @ Wres^T (shares the A fragment loads)
// Block = 256 threads = 8 waves; each wave owns a 16-row x 96-col strip.
// M/Wres staged once per block into LDS (padded stride K+1, odd -> conflict-
// free) via async-to-LDS; WMMAs feed from ds_load_b64. The same LDS buffer is
// reused (after a barrier) as the LayerNorm staging tile.
// ---------------------------------------------------------------------------
__global__ __launch_bounds__(256) void gemm_ln_res(
    const float* __restrict__ x,     // (8192, K)
    const float* __restrict__ M,     // (96, K)
    const float* __restrict__ Wres,  // (96, K) or nullptr
    const float* __restrict__ g,
    const float* __restrict__ b,
    int K,
    float* __restrict__ xn_out,      // (8192, 96)
    float* __restrict__ res_out)     // (8192, 96) or nullptr
{
    // Phase A: Ms[96*(K+1)] (+ Rs[96*(K+1)]) ; Phase B: xt tile [128*97]
    __shared__ float smem[12480];    // 49,920 B

    const int tid  = threadIdx.x;
    const int wave = tid >> 5;
    const int lane = tid & 31;
    const int hl   = lane & 15;          // row / col within 16-wide tile half
    const int kk   = (lane >> 4) * 2;    // 32-bit A/B frag: lanes 16-31 hold K=2,3
    const int row0 = blockIdx.x * 128 + wave * 16;
    const int KP   = K + 1;
    const bool doRes = (Wres != nullptr);

    // ---- stage B operands into LDS (async path, ASYNCcnt) ----
    float* Ms = smem;
    float* Rs = smem + D_ * KP;
    for (int idx = tid; idx < D_ * K; idx += 256) {
        const int i = idx / K, j = idx - i * K;
        async_ld_f32(&Ms[i * KP + j], &M[idx]);
    }
    if (doRes) {
        for (int idx = tid; idx < D_ * K; idx += 256) {
            const int i = idx / K, j = idx - i * K;
            async_ld_f32(&Rs[i * KP + j], &Wres[idx]);
        }
    }
    async_wait0();
    __syncthreads();

    v8f acc[6];
    v8f accR[6];
    for (int t = 0; t < 6; ++t) { acc[t] = {}; accR[t] = {}; }

    const float* arow = x + (size_t)(row0 + hl) * K;   // A rows stream from global

    for (int k0 = 0; k0 < K; k0 += 4) {
        // A fragment: A[m][k] = x[row0+m][k0+k]
        v2f a;
        a.x = arow[k0 + kk];
        a.y = arow[k0 + kk + 1];

        #pragma unroll
        for (int t = 0; t < 6; ++t) {
            const int n = t * 16 + hl;
            v2f bf;                         // B[k][n] = M[n][k0+k]  (xt = x @ M^T)
            bf.x = Ms[n * KP + k0 + kk];
            bf.y = Ms[n * KP + k0 + kk + 1];
            acc[t] = __builtin_amdgcn_wmma_f32_16x16x4_f32(
                false, a, false, bf, (short)0, acc[t], false, false);
        }
        if (doRes) {
            #pragma unroll
            for (int t = 0; t < 6; ++t) {
                const int n = t * 16 + hl;
                v2f bf;
                bf.x = Rs[n * KP + k0 + kk];
                bf.y = Rs[n * KP + k0 + kk + 1];
                accR[t] = __builtin_amdgcn_wmma_f32_16x16x4_f32(
                    false, a, false, bf, (short)0, accR[t], false, false);
            }
        }
    }

    // Residual needs no LN -> stream straight to global from D fragments.
    // D layout: VGPR r -> row r (lanes 0-15) / row r+8 (lanes 16-31), col = lane&15.
    const int radd = (lane >= 16) ? 8 : 0;
    if (doRes) {
        #pragma unroll
        for (int t = 0; t < 6; ++t)
            #pragma unroll
            for (int r = 0; r < 8; ++r) {
                const int grow = row0 + r + radd;
                res_out[grow * D_ + t * 16 + hl] = accR[t][r];
            }
    }

    __syncthreads();                 // all Ms/Rs reads done before smem reuse

    // Stage xt tile to LDS (stride 97) for the cross-column LayerNorm.
    #pragma unroll
    for (int t = 0; t < 6; ++t)
        #pragma unroll
        for (int r = 0; r < 8; ++r) {
            const int lrow = wave * 16 + r + radd;
            smem[lrow * 97 + t * 16 + hl] = acc[t][r];
        }
    __syncthreads();

    // 128 threads each LayerNorm one row of 96.
    if (tid < 128) {
        const float* rowp = &smem[tid * 97];
        float mu = 0.f;
        #pragma unroll 8
        for (int j = 0; j < D_; ++j) mu += rowp[j];
        mu *= (1.f / 96.f);
        float var = 0.f;
        #pragma unroll 8
        for (int j = 0; j < D_; ++j) { float d = rowp[j] - mu; var += d * d; }
        var *= (1.f / 96.f);
        const float rstd = rsqrtf(var + LN_EPS_F);
        const int grow = blockIdx.x * 128 + tid;
        #pragma unroll 8
        for (int j = 0; j < D_; ++j)
            xn_out[grow * D_ + j] = (rowp[j] - mu) * rstd * g[j] + b[j];
    }
}

// ---------------------------------------------------------------------------
// Kernel B: one block per sequence position s.
// Phase 1: W(s)[i][j] = sum_g P[i,j,g]*cos(2*pi*s/(i*768+j*8+g+2)) into LDS;
//          xn[b,s,:] batch rows async-copied into LDS (rows 8..15 zero).
// Phase 2: waves 0..5: out[b,s,i] = sum_j W[i][j]*xn[b,s,j] + res[b,s,i]
//          via fp32 WMMA: D(16i x 16b) = A(W tile 16x4) x B(xn 4x16).
// res may alias out (layer-2 identity residual, in-place add).
// ---------------------------------------------------------------------------
__global__ __launch_bounds__(256) void hier_matvec(
    const float* __restrict__ xn,   // (B,S,96)
    const float* __restrict__ P,    // (96,96,8)
    const float* __restrict__ res,  // (B,S,96)  (may alias out)
    float* __restrict__ out)        // (B,S,96)
{
    __shared__ float Wl[D_ * 100];  // W(s), stride 100 to avoid bank conflicts
    __shared__ float Xl[16 * D_];   // xn columns for 8 batches, rows 8..15 zero

    const int s   = blockIdx.x;
    const int tid = threadIdx.x;
    const float fs = (float)s;

    // Kick off the async xn -> LDS copies first; cos math hides their latency.
    for (int idx = tid; idx < 16 * D_; idx += 256) {
        const int bb = idx / D_;
        const int j  = idx - bb * D_;
        if (bb < 8) async_ld_f32(&Xl[idx], &xn[(bb * S_ + s) * D_ + j]);
        else        Xl[idx] = 0.f;
    }

    // Phase 1: cos-basis contraction (36 (i,j) pairs x 8 basis terms / thread).
    for (int idx = tid; idx < D_ * D_; idx += 256) {
        const int i = idx / D_;
        const int j = idx - i * D_;
        const float p0 = (float)(i * (D_ * NB_) + j * NB_ + 2);
        const float* pp = &P[idx * NB_];
        float w = 0.f;
        #pragma unroll
        for (int gg = 0; gg < NB_; ++gg)
            w += pp[gg] * __cosf(TWO_PI_F * fs / (p0 + (float)gg));
        Wl[i * 100 + j] = w;
    }
    async_wait0();
    __syncthreads();

    const int wave = tid >> 5;
    const int lane = tid & 31;
    if (wave < 6) {                      // uniform per-wave branch: EXEC all-1s
        const int hl = lane & 15;
        const int kk = (lane >> 4) * 2;
        v8f acc = {};
        for (int k0 = 0; k0 < D_; k0 += 4) {
            v2f a, bf;
            a.x  = Wl[(wave * 16 + hl) * 100 + k0 + kk];      // A[m][k] = W[i0+m][k]
            a.y  = Wl[(wave * 16 + hl) * 100 + k0 + kk + 1];
            bf.x = Xl[hl * D_ + k0 + kk];                     // B[k][n] = xn[b=n][k]
            bf.y = Xl[hl * D_ + k0 + kk + 1];
            acc = __builtin_amdgcn_wmma_f32_16x16x4_f32(
                false, a, false, bf, (short)0, acc, false, false);
        }
        const int bb = hl;               // batch column of D tile
        if (bb < 8) {
            const int radd = (lane >= 16) ? 8 : 0;
            #pragma unroll
            for (int r = 0; r < 8; ++r) {
                const int i   = wave * 16 + r + radd;
                const int off = (bb * S_ + s) * D_ + i;
                out[off] = acc[r] + res[off];
            }
        }
    }
}

extern "C" void kernel_launch(void* const* d_in, const int* in_sizes, int n_in,
                              void* d_out, int out_size, void* d_ws, size_t ws_size,
                              hipStream_t stream) {
    (void)in_sizes; (void)n_in; (void)out_size; (void)ws_size;
    const float* seq   = (const float*)d_in[0];
    const float* M1    = (const float*)d_in[1];
    const float* P1    = (const float*)d_in[2];
    const float* Wres1 = (const float*)d_in[3];
    const float* g1    = (const float*)d_in[4];
    const float* b1    = (const float*)d_in[5];
    const float* M2    = (const float*)d_in[6];
    const float* P2    = (const float*)d_in[7];
    const float* g2    = (const float*)d_in[8];
    const float* b2    = (const float*)d_in[9];

    float* out = (float*)d_out;                      // holds x1, then final x2
    float* xn  = (float*)d_ws;                       // xn1 then reused for xn2
    float* r1  = xn + (size_t)B_ * S_ * D_;          // layer-1 projected residual
    // ws usage: 2 * 8*1024*96 * 4B = 6.3 MB

    // Layer 1: 64 -> 96
    gemm_ln_res<<<64, 256, 0, stream>>>(seq, M1, Wres1, g1, b1, DIN_, xn, r1);
    hier_matvec<<<S_, 256, 0, stream>>>(xn, P1, r1, out);        // out = x1

    // Layer 2: 96 -> 96, identity residual (in-place on d_out)
    gemm_ln_res<<<64, 256, 0, stream>>>(out, M2, nullptr, g2, b2, D_, xn, nullptr);
    hier_matvec<<<S_, 256, 0, stream>>>(xn, P2, out, out);       // out = x2
}